// LocalRandomMean_35407710388372
// MI455X (gfx1250) — compile-verified
//
#include <hip/hip_runtime.h>

typedef unsigned int uint32;
typedef float v2f __attribute__((ext_vector_type(2)));
typedef float v8f __attribute__((ext_vector_type(8)));

#define NCELLS   393216   // 32*3*64*64
#define NBLOCKS  1536     // NCELLS / 256
#define KEYMASK64 0xFFFFFFC0u
#define KEYMASK16 0xFFFFFFF0u

// Index-salted sortable key: noise in [0,1) -> positive float -> u32-monotone.
// Low bits replaced by element index => unique keys, exact tie-break by index.
__device__ __forceinline__ uint32 fkey(float v, int idx, uint32 mask) {
  return (__float_as_uint(v) & mask) | (uint32)idx;
}

__device__ __forceinline__ uint32 umin32(uint32 a, uint32 b) { return a < b ? a : b; }
__device__ __forceinline__ uint32 umax32(uint32 a, uint32 b) { return a < b ? b : a; }

// Fully-unrolled ascending bitonic sort on N u32 keys held in registers.
template <int N>
__device__ __forceinline__ void bitonic_sort(uint32* a) {
#pragma unroll
  for (int k = 2; k <= N; k <<= 1) {
#pragma unroll
    for (int j = k >> 1; j > 0; j >>= 1) {
#pragma unroll
      for (int i = 0; i < N; ++i) {
        const int l = i ^ j;
        if (l > i) {
          const uint32 u = a[i], v = a[l];
          const uint32 lo = umin32(u, v);    // v_min_u32
          const uint32 hi = umax32(u, v);    // v_max_u32
          const bool up = ((i & k) == 0);    // compile-time constant
          a[i] = up ? lo : hi;
          a[l] = up ? hi : lo;
        }
      }
    }
  }
}

// Rank-(N/2) element (0-indexed) of a[0..N-1], i.e. the (N/2+1)-th smallest:
// sort the two halves, then T = min_i max(A[i], B[N/2-1-i]).
// (Bitonic-split: {max(A[i],B[H-1-i])} is exactly the multiset of the H largest.)
template <int N>
__device__ __forceinline__ uint32 select_half(uint32* a) {
  constexpr int H = N / 2;
  bitonic_sort<H>(a);
  bitonic_sort<H>(a + H);
  uint32 u[H];
#pragma unroll
  for (int i = 0; i < H; ++i) u[i] = umax32(a[i], a[N - 1 - i]);
#pragma unroll
  for (int s = H / 2; s > 0; s >>= 1) {
#pragma unroll
    for (int i = 0; i < s; ++i) u[i] = umin32(u[i], u[i + s]);  // min tree (ILP)
  }
  return u[0];
}

__global__ __launch_bounds__(256)
void lrm_main(const float* __restrict__ x, const float* __restrict__ y,
              const float* __restrict__ nx, const float* __restrict__ ny,
              float* __restrict__ blocksum) {
  const int m  = blockIdx.x * 256 + threadIdx.x;   // cell id
  const int wc = m & 63;                           // window col
  const int r  = (m >> 6) & 63;                    // window row
  const int bc = m >> 12;                          // fused batch*channel

  const size_t xoff = ((size_t)bc << 18) + (size_t)(r << 3) * 512 + (size_t)(wc << 3);
  const size_t yoff = ((size_t)bc << 16) + (size_t)(r << 2) * 256 + (size_t)(wc << 2);
  const float* xp  = x  + xoff;
  const float* nxp = nx + xoff;
  const float* yp  = y  + yoff;
  const float* nyp = ny + yoff;

  // ---------------- x: 8x8 window, keep the 32 smallest-noise elements ------
  uint32 a[64];
#pragma unroll
  for (int rr = 0; rr < 8; ++rr) {
    const float4 n0 = *(const float4*)(nxp + rr * 512);
    const float4 n1 = *(const float4*)(nxp + rr * 512 + 4);
    const int b = rr * 8;
    a[b + 0] = fkey(n0.x, b + 0, KEYMASK64);
    a[b + 1] = fkey(n0.y, b + 1, KEYMASK64);
    a[b + 2] = fkey(n0.z, b + 2, KEYMASK64);
    a[b + 3] = fkey(n0.w, b + 3, KEYMASK64);
    a[b + 4] = fkey(n1.x, b + 4, KEYMASK64);
    a[b + 5] = fkey(n1.y, b + 5, KEYMASK64);
    a[b + 6] = fkey(n1.z, b + 6, KEYMASK64);
    a[b + 7] = fkey(n1.w, b + 7, KEYMASK64);
  }
  const uint32 Tx = select_half<64>(a);   // keep iff key < Tx (exactly 32 kept)

  float sx = 0.f;
#pragma unroll
  for (int rr = 0; rr < 8; ++rr) {
    const float4 n0 = *(const float4*)(nxp + rr * 512);      // CSE / cache hit
    const float4 n1 = *(const float4*)(nxp + rr * 512 + 4);
    const float4 x0 = *(const float4*)(xp + rr * 512);
    const float4 x1 = *(const float4*)(xp + rr * 512 + 4);
    const int b = rr * 8;
    sx += (fkey(n0.x, b + 0, KEYMASK64) < Tx) ? x0.x : 0.f;
    sx += (fkey(n0.y, b + 1, KEYMASK64) < Tx) ? x0.y : 0.f;
    sx += (fkey(n0.z, b + 2, KEYMASK64) < Tx) ? x0.z : 0.f;
    sx += (fkey(n0.w, b + 3, KEYMASK64) < Tx) ? x0.w : 0.f;
    sx += (fkey(n1.x, b + 4, KEYMASK64) < Tx) ? x1.x : 0.f;
    sx += (fkey(n1.y, b + 5, KEYMASK64) < Tx) ? x1.y : 0.f;
    sx += (fkey(n1.z, b + 6, KEYMASK64) < Tx) ? x1.z : 0.f;
    sx += (fkey(n1.w, b + 7, KEYMASK64) < Tx) ? x1.w : 0.f;
  }
  // mean(mask*x) over 64 elems, mask = 2 on kept: (2*sx)/64
  const float xpool = sx * (1.0f / 32.0f);

  // ---------------- y: 4x4 window, keep the 8 smallest-noise elements -------
  uint32 c[16];
#pragma unroll
  for (int rr = 0; rr < 4; ++rr) {
    const float4 n0 = *(const float4*)(nyp + rr * 256);
    const int b = rr * 4;
    c[b + 0] = fkey(n0.x, b + 0, KEYMASK16);
    c[b + 1] = fkey(n0.y, b + 1, KEYMASK16);
    c[b + 2] = fkey(n0.z, b + 2, KEYMASK16);
    c[b + 3] = fkey(n0.w, b + 3, KEYMASK16);
  }
  const uint32 Ty = select_half<16>(c);

  float sy = 0.f;
#pragma unroll
  for (int rr = 0; rr < 4; ++rr) {
    const float4 n0 = *(const float4*)(nyp + rr * 256);
    const float4 y0 = *(const float4*)(yp + rr * 256);
    const int b = rr * 4;
    sy += (fkey(n0.x, b + 0, KEYMASK16) < Ty) ? y0.x : 0.f;
    sy += (fkey(n0.y, b + 1, KEYMASK16) < Ty) ? y0.y : 0.f;
    sy += (fkey(n0.z, b + 2, KEYMASK16) < Ty) ? y0.z : 0.f;
    sy += (fkey(n0.w, b + 3, KEYMASK16) < Ty) ? y0.w : 0.f;
  }
  const float ypool = sy * (1.0f / 8.0f);

  const float d = xpool - ypool;
  const float partial = d * d * (1.0f / (float)NCELLS);

  // ------------- wave reduction: fp32 WMMA (A=ones) folds lane n with n+16 --
  // D[m][n] = sum_k B[k][n]; B[0][n]=partial(lane n), B[2][n]=partial(lane n+16).
  v2f A;  A.x = 1.0f;     A.y = 1.0f;
  v2f Bv; Bv.x = partial; Bv.y = 0.0f;
  v8f Cv = {};
  Cv = __builtin_amdgcn_wmma_f32_16x16x4_f32(false, A, false, Bv,
                                             (short)0, Cv, false, false);
  float s = Cv[0];                       // lane j holds p_{j&15} + p_{(j&15)+16}
  s += __shfl_xor(s, 1);
  s += __shfl_xor(s, 2);
  s += __shfl_xor(s, 4);
  s += __shfl_xor(s, 8);                 // full 32-lane sum in all lanes

  __shared__ float red[8];
  if ((threadIdx.x & 31) == 0) red[threadIdx.x >> 5] = s;
  __syncthreads();
  if (threadIdx.x == 0) {
    float t = 0.f;
#pragma unroll
    for (int i = 0; i < 8; ++i) t += red[i];
    blocksum[blockIdx.x] = t;
  }
}

// Deterministic fixed-order final reduction (no float atomics).
__global__ __launch_bounds__(256)
void lrm_reduce(const float* __restrict__ bs, float* __restrict__ out, int n) {
  __shared__ float sm[256];
  float t = 0.f;
  for (int i = threadIdx.x; i < n; i += 256) t += bs[i];
  sm[threadIdx.x] = t;
  __syncthreads();
  for (int s2 = 128; s2 > 0; s2 >>= 1) {
    if ((int)threadIdx.x < s2) sm[threadIdx.x] += sm[threadIdx.x + s2];
    __syncthreads();
  }
  if (threadIdx.x == 0) out[0] = sm[0];
}

extern "C" void kernel_launch(void* const* d_in, const int* in_sizes, int n_in,
                              void* d_out, int out_size, void* d_ws, size_t ws_size,
                              hipStream_t stream) {
  const float* x  = (const float*)d_in[0];   // [32,3,512,512]
  const float* y  = (const float*)d_in[1];   // [32,3,256,256]
  const float* nx = (const float*)d_in[2];
  const float* ny = (const float*)d_in[3];
  float* bs = (float*)d_ws;                  // NBLOCKS floats of scratch

  lrm_main<<<NBLOCKS, 256, 0, stream>>>(x, y, nx, ny, bs);
  lrm_reduce<<<1, 256, 0, stream>>>(bs, (float*)d_out, NBLOCKS);
}